// LightGCN_31602369364530
// MI455X (gfx1250) — compile-verified
//
#include <hip/hip_runtime.h>

// ---------------------------------------------------------------------------
// LightGCN forward on MI455X (gfx1250).
// 3x SpMM (COO, E=1.2M) over [150000 x 64] fp32 + elementwise accumulate.
// AI ~0.24 flop/byte -> pure bandwidth problem; whole working set (~130 MB)
// is resident in the 192 MB global L2.
// wave32: each wave processes a batch of 32 edges. Index triplets are staged
// into LDS with global_load_async_to_lds_b32 (ASYNCcnt, per-wave wait), then
// broadcast per edge with v_readlane. Gather = 256B coalesced global_load_b64
// per wave; scatter = non-returning global_atomic_add_f32 resolved in L2.
// ---------------------------------------------------------------------------

namespace {
constexpr int kUsers    = 100000;
constexpr int kItems    = 50000;
constexpr int kNodes    = kUsers + kItems;   // 150000
constexpr int kD        = 64;
constexpr int kE        = 1200000;           // multiple of 32 -> full waves
constexpr int kQuad     = kNodes * kD / 4;   // float4 elements per node matrix
constexpr int kUserQuad = kUsers * kD / 4;
}  // namespace

// acc = h0 = concat(user_emb, item_emb); h1 = 0 (ready for layer 0 scatter)
__global__ void __launch_bounds__(256) lgcn_init(
    const float4* __restrict__ user4, const float4* __restrict__ item4,
    float4* __restrict__ acc4, float4* __restrict__ h04,
    float4* __restrict__ h14) {
  int i = blockIdx.x * 256 + threadIdx.x;
  if (i >= kQuad) return;
  const float4 v = (i < kUserQuad) ? user4[i] : item4[i - kUserQuad];
  acc4[i] = v;
  h04[i]  = v;
  h14[i]  = make_float4(0.f, 0.f, 0.f, 0.f);
}

// y[rows[e]] += vals[e] * x[cols[e]]  -- one wave32 per 32-edge batch.
__global__ void __launch_bounds__(256) lgcn_spmm(
    const int* __restrict__ rows, const int* __restrict__ cols,
    const float* __restrict__ vals,
    const float* __restrict__ x, float* __restrict__ y) {
  __shared__ int   s_r[256];
  __shared__ int   s_c[256];
  __shared__ float s_v[256];

  const int tid  = blockIdx.x * 256 + threadIdx.x;  // == staging edge id
  const int lane = threadIdx.x & 31;
  if (tid >= kE) return;  // kE % 32 == 0 -> whole waves exit uniformly

  // gfx1250 global_prefetch_b8: pull upcoming index cachelines toward L2/L0.
  if (lane == 0 && (tid + 16384) < kE) {
    __builtin_prefetch(cols + tid + 16384, 0, 0);
    __builtin_prefetch(rows + tid + 16384, 0, 0);
    __builtin_prefetch(vals + tid + 16384, 0, 0);
  }

  // ---- Async staging of this wave's 32 (row, col, val) triplets into LDS.
  // GVS mode: SGPR base + per-lane byte offset. ASYNCcnt is per-wave, so a
  // s_wait_asynccnt 0 (no barrier) orders the staging against the ds_loads.
  {
    const unsigned voff = (unsigned)tid * 4u;  // byte offset into index arrays
    const unsigned aR = (unsigned)(uintptr_t)&s_r[threadIdx.x];
    const unsigned aC = (unsigned)(uintptr_t)&s_c[threadIdx.x];
    const unsigned aV = (unsigned)(uintptr_t)&s_v[threadIdx.x];
    asm volatile("global_load_async_to_lds_b32 %0, %1, %2"
                 :: "v"(aR), "v"(voff), "s"(rows) : "memory");
    asm volatile("global_load_async_to_lds_b32 %0, %1, %2"
                 :: "v"(aC), "v"(voff), "s"(cols) : "memory");
    asm volatile("global_load_async_to_lds_b32 %0, %1, %2"
                 :: "v"(aV), "v"(voff), "s"(vals) : "memory");
    asm volatile("s_wait_asynccnt 0x0" ::: "memory");
  }

  // Coalesced read-back: lane i holds edge (waveBase + i)'s triplet.
  const int   rv = s_r[threadIdx.x];
  const int   cv = s_c[threadIdx.x];
  const int   vb = __builtin_bit_cast(int, s_v[threadIdx.x]);

  // ---- Process the 32 edges; broadcast indices with v_readlane.
#pragma unroll 4
  for (int i = 0; i < 32; ++i) {
    const int   c = __builtin_amdgcn_readlane(cv, i);
    const int   r = __builtin_amdgcn_readlane(rv, i);
    const float v = __builtin_bit_cast(float, __builtin_amdgcn_readlane(vb, i));

    // 256B coalesced row gather (L2-resident): global_load_b64 per lane.
    const float2 g = *reinterpret_cast<const float2*>(
        x + (size_t)(unsigned)c * kD + lane * 2);

    // Scatter via hardware f32 atomics in L2 (agent scope, non-returning).
    float* yr = y + (size_t)(unsigned)r * kD + lane * 2;
    __hip_atomic_fetch_add(yr + 0, g.x * v, __ATOMIC_RELAXED,
                           __HIP_MEMORY_SCOPE_AGENT);
    __hip_atomic_fetch_add(yr + 1, g.y * v, __ATOMIC_RELAXED,
                           __HIP_MEMORY_SCOPE_AGENT);
  }
}

// acc += h; z = 0  (z is the just-consumed ping buffer, safe to clear here)
__global__ void __launch_bounds__(256) lgcn_acc_zero(
    float4* __restrict__ acc4, const float4* __restrict__ h4,
    float4* __restrict__ z4) {
  int i = blockIdx.x * 256 + threadIdx.x;
  if (i >= kQuad) return;
  float4 a = acc4[i];
  const float4 h = h4[i];
  a.x += h.x; a.y += h.y; a.z += h.z; a.w += h.w;
  acc4[i] = a;
  z4[i] = make_float4(0.f, 0.f, 0.f, 0.f);
}

// Last layer: fold the final accumulate with the 1/(L+1) scale.
__global__ void __launch_bounds__(256) lgcn_final(
    float4* __restrict__ acc4, const float4* __restrict__ h4) {
  int i = blockIdx.x * 256 + threadIdx.x;
  if (i >= kQuad) return;
  float4 a = acc4[i];
  const float4 h = h4[i];
  constexpr float s = 0.25f;  // 1 / (L + 1)
  a.x = (a.x + h.x) * s;
  a.y = (a.y + h.y) * s;
  a.z = (a.z + h.z) * s;
  a.w = (a.w + h.w) * s;
  acc4[i] = a;
}

extern "C" void kernel_launch(void* const* d_in, const int* in_sizes, int n_in,
                              void* d_out, int out_size, void* d_ws,
                              size_t ws_size, hipStream_t stream) {
  (void)in_sizes; (void)n_in; (void)out_size; (void)ws_size;

  const float4* user4 = (const float4*)d_in[0];
  const float4* item4 = (const float4*)d_in[1];
  const int*    rows  = (const int*)d_in[2];
  const int*    cols  = (const int*)d_in[3];
  const float*  vals  = (const float*)d_in[4];

  float* acc = (float*)d_out;                 // [N, D] accumulator == output
  float* h0  = (float*)d_ws;                  // ping
  float* h1  = h0 + (size_t)kNodes * kD;      // pong

  const int qBlocks = (kQuad + 255) / 256;
  const int sBlocks = (kE + 255) / 256;       // one thread per staged edge

  // init: acc = h0 = x, h1 = 0
  lgcn_init<<<qBlocks, 256, 0, stream>>>(user4, item4, (float4*)acc,
                                         (float4*)h0, (float4*)h1);

  // L0: h1 = A*h0 ; acc += h1 ; h0 = 0
  lgcn_spmm<<<sBlocks, 256, 0, stream>>>(rows, cols, vals, h0, h1);
  lgcn_acc_zero<<<qBlocks, 256, 0, stream>>>((float4*)acc, (const float4*)h1,
                                             (float4*)h0);

  // L1: h0 = A*h1 ; acc += h0 ; h1 = 0
  lgcn_spmm<<<sBlocks, 256, 0, stream>>>(rows, cols, vals, h1, h0);
  lgcn_acc_zero<<<qBlocks, 256, 0, stream>>>((float4*)acc, (const float4*)h0,
                                             (float4*)h1);

  // L2: h1 = A*h0 ; acc = (acc + h1) / 4
  lgcn_spmm<<<sBlocks, 256, 0, stream>>>(rows, cols, vals, h0, h1);
  lgcn_final<<<qBlocks, 256, 0, stream>>>((float4*)acc, (const float4*)h1);
}